// TimeAttentionBlock_82643760710000
// MI455X (gfx1250) — compile-verified
//
#include <hip/hip_runtime.h>
#include <hip/hip_bf16.h>

// ---------------------------------------------------------------------------
// CDNA5 (gfx1250) implementation of the TimeAttentionBlock.
// Algebraic collapse: entire block == relu(R_b @ x_b + k) where R_b is a
// per-batch 64x64 matrix derived from the Gram matrix G_b = x_b x_b^T and the
// exact BN(train-mode) moments, which are quadratic in x and therefore
// recoverable from G_b and the column sums sx_b.
// Heavy passes (Gram + apply) and the per-batch attention solve all use
// V_WMMA_F32_16X16X4_F32; X tiles are DMA'd into LDS by the Tensor Data
// Mover (tensor_load_to_lds + s_wait_tensorcnt) when the builtin exists.
// ---------------------------------------------------------------------------

typedef float v2f __attribute__((ext_vector_type(2)));
typedef float v8f __attribute__((ext_vector_type(8)));
typedef unsigned int u32x4 __attribute__((ext_vector_type(4)));
typedef int i32x4 __attribute__((ext_vector_type(4)));
typedef int i32x8 __attribute__((ext_vector_type(8)));

#define XS 257      // LDS row stride for 256-wide x tiles (bank-conflict pad)
#define LSTRIDE 8192
#define NB 16       // batches
#define CIN 64
#define HC 256
#define BLF (1.0f/131072.0f)   // 1/(B*L)
#define EPS_ 1e-5f

#if defined(__gfx1250__) && __has_builtin(__builtin_amdgcn_tensor_load_to_lds) && \
    __has_builtin(__builtin_amdgcn_s_wait_tensorcnt)
#define HAVE_TDM 1
#else
#define HAVE_TDM 0
#endif

__device__ __forceinline__ v8f wmma4(v2f a, v2f b, v8f c) {
#if __has_builtin(__builtin_amdgcn_wmma_f32_16x16x4_f32)
  // D = A(16x4 f32) * B(4x16 f32) + C(16x16 f32)
  return __builtin_amdgcn_wmma_f32_16x16x4_f32(false, a, false, b, (short)0, c,
                                               false, false);
#else
  c[0] += a[0] * b[0] + a[1] * b[1];
  return c;
#endif
}

#if HAVE_TDM
// DMA a 64-row x 256-f32 tile (row stride 8192 f32 in memory) into LDS offset
// 0, with a 1-DWORD pad inserted after every 256 DWORDs so the LDS layout is
// exactly the 257-float padded stride the WMMA fragment readers expect.
__device__ __forceinline__ void tdm_load_x_tile(const float* gptr) {
  unsigned long long ga = (unsigned long long)(uintptr_t)gptr;
  u32x4 g0;
  g0[0] = 1u;                                  // count=1 (valid user D#)
  g0[1] = 0u;                                  // lds_addr = 0 (tile is first in dynamic LDS)
  g0[2] = (unsigned)(ga & 0xffffffffu);        // global_addr[31:0]
  g0[3] = (unsigned)((ga >> 32) & 0x01ffffffu) // global_addr[56:32]
        | (2u << 30);                          // type=2 ("image")
  i32x8 g1;
  g1[0] = (2 << 16)       // data_size = 4 bytes
        | (1 << 20)       // pad_enable
        | (7 << 22)       // pad_interval = 256 DWORDs
        | (0 << 25);      // pad_amount  = 1 DWORD  -> LDS row stride 257 f32
  g1[1] = (int)((8192u & 0xffffu) << 16);      // tensor_dim0[15:0] (=8192)
  g1[2] = (int)((8192u >> 16) | (64u << 16));  // tensor_dim0[31:16], tensor_dim1[15:0] (=64)
  g1[3] = (int)((0u) | (256u << 16));          // tensor_dim1[31:16], tile_dim0=256
  g1[4] = (int)(64u);                          // tile_dim1=64, tile_dim2=0
  g1[5] = (int)8192;                           // tensor_dim0_stride[31:0] (=8192 elems)
  g1[6] = 0;                                   // stride0[47:32], stride1[15:0]
  g1[7] = 0;
  i32x4 g2 = {0, 0, 0, 0};
  i32x4 g3 = {0, 0, 0, 0};
#if __clang_major__ >= 23
  i32x8 g1b = {0, 0, 0, 0, 0, 0, 0, 0};
  __builtin_amdgcn_tensor_load_to_lds(g0, g1, g2, g3, g1b, 0);
#else
  __builtin_amdgcn_tensor_load_to_lds(g0, g1, g2, g3, 0);
#endif
}
#endif

// --------------------------------------------------------------------------
__global__ void k_zero(float* __restrict__ p, int n) {
  int i = blockIdx.x * 256 + threadIdx.x;
  if (i < n) p[i] = 0.0f;
}

// --------------------------------------------------------------------------
// Pass 1: G_b += X_chunk X_chunk^T (64x64), SX_b += row sums.
// grid = 16 batches * 32 chunks of 256, block = 256 (8 waves).
__global__ void k_gram(const float* __restrict__ x, float* __restrict__ G,
                       float* __restrict__ SX) {
  extern __shared__ float sm[];
  float* lx = sm;  // 64 x XS  (MUST stay first: TDM writes LDS offset 0)
  const int t = threadIdx.x;
  const int b = blockIdx.x >> 5;
  const int l0 = (blockIdx.x & 31) * 256;
  const float* xb = x + ((size_t)b * CIN) * LSTRIDE + l0;

#if HAVE_TDM
  if (t == 0) {
    tdm_load_x_tile(xb);
    __builtin_amdgcn_s_wait_tensorcnt(0);
  }
#else
  for (int i = t; i < 64 * 64; i += 256) {  // 64 rows x 64 float4
    int row = i >> 6, c4 = i & 63;
    float4 v = reinterpret_cast<const float4*>(xb + (size_t)row * LSTRIDE)[c4];
    float* d = lx + row * XS + c4 * 4;
    d[0] = v.x; d[1] = v.y; d[2] = v.z; d[3] = v.w;
  }
#endif
  __syncthreads();

  // per-channel partial column sums (4 segments of 64 per channel)
  {
    int c = t & 63, p4 = t >> 6;
    const float* r = lx + c * XS + p4 * 64;
    float s = 0.0f;
    for (int l = 0; l < 64; ++l) s += r[l];
    atomicAdd(&SX[b * CIN + c], s);
  }

  const int wave = t >> 5, lane = t & 31;
  const int half = lane >> 4, ln = lane & 15;
  const int jt = wave & 3;        // j-tile 0..3
  const int i1 = wave >> 2;       // i-tile 0..1
  const int i2 = i1 + 2;          // i-tile 2..3
  v8f acc1 = {}; v8f acc2 = {};
  const float* ra1 = lx + (i1 * 16 + ln) * XS;
  const float* ra2 = lx + (i2 * 16 + ln) * XS;
  const float* rb  = lx + (jt * 16 + ln) * XS;
  for (int k0 = 0; k0 < 256; k0 += 4) {
    const int k = k0 + 2 * half;
    v2f a1; a1[0] = ra1[k]; a1[1] = ra1[k + 1];
    v2f a2; a2[0] = ra2[k]; a2[1] = ra2[k + 1];
    v2f bb; bb[0] = rb[k];  bb[1] = rb[k + 1];
    acc1 = wmma4(a1, bb, acc1);
    acc2 = wmma4(a2, bb, acc2);
  }
  float* Gb = G + b * 4096;
  const int mb = half * 8;
  for (int r = 0; r < 8; ++r) {
    atomicAdd(&Gb[(i1 * 16 + mb + r) * 64 + jt * 16 + ln], acc1[r]);
    atomicAdd(&Gb[(i2 * 16 + mb + r) * 64 + jt * 16 + ln], acc2[r]);
  }
}

// --------------------------------------------------------------------------
// T_b = Wq(256x64) @ G_b(64x64). grid = 16, block = 256.
__global__ void k_tmat(const float* __restrict__ wq, const float* __restrict__ G,
                       float* __restrict__ T) {
  extern __shared__ float sm[];
  float* lwq = sm;              // 256 x 65
  float* lg  = lwq + 256 * 65;  // 64 x 65
  const int t = threadIdx.x, b = blockIdx.x;
  for (int i = t; i < HC * CIN; i += 256) lwq[(i >> 6) * 65 + (i & 63)] = wq[i];
  for (int i = t; i < 64 * 64; i += 256)  lg[(i >> 6) * 65 + (i & 63)] = G[b * 4096 + i];
  __syncthreads();
  const int wave = t >> 5, lane = t & 31, half = lane >> 4, ln = lane & 15;
  v8f acc[2][4] = {};
  for (int k0 = 0; k0 < 64; k0 += 4) {
    const int k = k0 + 2 * half;
    const float* r1 = lwq + (wave * 16 + ln) * 65;
    const float* r2 = lwq + ((wave + 8) * 16 + ln) * 65;
    v2f a1; a1[0] = r1[k]; a1[1] = r1[k + 1];
    v2f a2; a2[0] = r2[k]; a2[1] = r2[k + 1];
    for (int n = 0; n < 4; ++n) {
      v2f bb; bb[0] = lg[k * 65 + n * 16 + ln]; bb[1] = lg[(k + 1) * 65 + n * 16 + ln];
      acc[0][n] = wmma4(a1, bb, acc[0][n]);
      acc[1][n] = wmma4(a2, bb, acc[1][n]);
    }
  }
  float* Tb = T + b * HC * CIN;
  for (int it = 0; it < 2; ++it)
    for (int n = 0; n < 4; ++n)
      for (int r = 0; r < 8; ++r) {
        int row = (wave + 8 * it) * 16 + half * 8 + r;
        Tb[row * 64 + n * 16 + ln] = acc[it][n][r];
      }
}

// --------------------------------------------------------------------------
// Per (b, 16-column j-block): S = T_b @ Wk[j]^T, column softmax (axis=i),
// M_b += exp(S-max)/den/sqrt(8) @ Wv[j].  grid = 16*16, block = 256.
__global__ void k_scores(const float* __restrict__ wk, const float* __restrict__ wv,
                         const float* __restrict__ T, float* __restrict__ M) {
  extern __shared__ float sm[];
  float* lt   = sm;                // 256 x 65
  float* lwk  = lt + 256 * 65;     // 16 x 65
  float* sc   = lwk + 16 * 65;     // 256 x 17
  float* psum = sc + 256 * 17;     // 16 x 17 (reused: partial max, partial sum)
  float* mx   = psum + 16 * 17;    // 16
  float* fden = mx + 16;           // 16
  float* lwvs = fden + 16;         // 16 x 65
  const int t = threadIdx.x;
  const int b = blockIdx.x >> 4;
  const int j0 = (blockIdx.x & 15) * 16;
  const float* Tb = T + b * HC * CIN;
  for (int i = t; i < HC * CIN; i += 256) lt[(i >> 6) * 65 + (i & 63)] = Tb[i];
  for (int i = t; i < 16 * 64; i += 256)
    lwk[(i >> 6) * 65 + (i & 63)] = wk[(j0 + (i >> 6)) * 64 + (i & 63)];
  __syncthreads();
  const int wave = t >> 5, lane = t & 31, half = lane >> 4, ln = lane & 15;
  v8f sa = {}; v8f sb = {};
  for (int k0 = 0; k0 < 64; k0 += 4) {
    const int k = k0 + 2 * half;
    const float* r1 = lt + (wave * 16 + ln) * 65;
    const float* r2 = lt + ((wave + 8) * 16 + ln) * 65;
    v2f a1; a1[0] = r1[k]; a1[1] = r1[k + 1];
    v2f a2; a2[0] = r2[k]; a2[1] = r2[k + 1];
    v2f bb; bb[0] = lwk[ln * 65 + k]; bb[1] = lwk[ln * 65 + k + 1]; // B = Wk^T
    sa = wmma4(a1, bb, sa);
    sb = wmma4(a2, bb, sb);
  }
  for (int r = 0; r < 8; ++r) {
    sc[(wave * 16 + half * 8 + r) * 17 + ln] = sa[r];
    sc[((wave + 8) * 16 + half * 8 + r) * 17 + ln] = sb[r];
  }
  __syncthreads();
  {  // two-level column max over i (axis=1 softmax)
    const int jj = t & 15, seg = t >> 4;
    float m = -3.4e38f;
    for (int i = seg * 16; i < seg * 16 + 16; ++i) m = fmaxf(m, sc[i * 17 + jj]);
    psum[jj * 17 + seg] = m;
  }
  __syncthreads();
  if (t < 16) {
    float m = -3.4e38f;
    for (int s = 0; s < 16; ++s) m = fmaxf(m, psum[t * 17 + s]);
    mx[t] = m;
  }
  __syncthreads();
  {  // exponentiate + partial column sums
    const int jj = t & 15, seg = t >> 4;
    const float m = mx[jj];
    float s = 0.0f;
    for (int i = seg * 16; i < seg * 16 + 16; ++i) {
      float e = __expf(sc[i * 17 + jj] - m);
      sc[i * 17 + jj] = e;
      s += e;
    }
    psum[jj * 17 + seg] = s;
  }
  __syncthreads();
  if (t < 16) {
    float d = 0.0f;
    for (int s = 0; s < 16; ++s) d += psum[t * 17 + s];
    fden[t] = 1.0f / (d * 2.8284271247461903f);  // fold 1/den and 1/sqrt(8)
  }
  __syncthreads();
  for (int i = t; i < 16 * 64; i += 256) {
    int jj = i >> 6, c = i & 63;
    lwvs[jj * 65 + c] = wv[(j0 + jj) * 64 + c] * fden[jj];
  }
  __syncthreads();
  v8f macc[2][4] = {};
  for (int k0 = 0; k0 < 16; k0 += 4) {
    const int k = k0 + 2 * half;
    const float* r1 = sc + (wave * 16 + ln) * 17;
    const float* r2 = sc + ((wave + 8) * 16 + ln) * 17;
    v2f a1; a1[0] = r1[k]; a1[1] = r1[k + 1];
    v2f a2; a2[0] = r2[k]; a2[1] = r2[k + 1];
    for (int n = 0; n < 4; ++n) {
      v2f bb; bb[0] = lwvs[k * 65 + n * 16 + ln]; bb[1] = lwvs[(k + 1) * 65 + n * 16 + ln];
      macc[0][n] = wmma4(a1, bb, macc[0][n]);
      macc[1][n] = wmma4(a2, bb, macc[1][n]);
    }
  }
  float* Mb = M + b * HC * CIN;
  for (int it = 0; it < 2; ++it)
    for (int n = 0; n < 4; ++n)
      for (int r = 0; r < 8; ++r) {
        int row = (wave + 8 * it) * 16 + half * 8 + r;
        atomicAdd(&Mb[row * 64 + n * 16 + ln], macc[it][n][r]);
      }
}

// --------------------------------------------------------------------------
// P_b = Wc(64x256) @ M_b(256x64);  ysum_b = P sx;  yssum_b[c] = p_c^T G p_c.
__global__ void k_pmat(const float* __restrict__ wc, const float* __restrict__ M,
                       const float* __restrict__ G, const float* __restrict__ SX,
                       float* __restrict__ P, float* __restrict__ ysum,
                       float* __restrict__ yssum) {
  extern __shared__ float sm[];
  float* lwc = sm;               // 64 x 257
  float* lm  = lwc + 64 * 257;   // 256 x 65
  float* lg  = lm + 256 * 65;    // 64 x 65
  float* lp  = lg + 64 * 65;     // 64 x 65
  float* lsx = lp + 64 * 65;     // 64
  const int t = threadIdx.x, b = blockIdx.x;
  for (int i = t; i < 64 * 256; i += 256) lwc[(i >> 8) * 257 + (i & 255)] = wc[i];
  const float* Mb = M + b * HC * CIN;
  for (int i = t; i < 256 * 64; i += 256) lm[(i >> 6) * 65 + (i & 63)] = Mb[i];
  for (int i = t; i < 64 * 64; i += 256)  lg[(i >> 6) * 65 + (i & 63)] = G[b * 4096 + i];
  if (t < 64) lsx[t] = SX[b * 64 + t];
  __syncthreads();
  const int wave = t >> 5, lane = t & 31, half = lane >> 4, ln = lane & 15;
  const int i1 = wave >> 2, j = wave & 3, i2 = i1 + 2;
  v8f acc1 = {}; v8f acc2 = {};
  for (int k0 = 0; k0 < 256; k0 += 4) {
    const int k = k0 + 2 * half;
    const float* r1 = lwc + (i1 * 16 + ln) * 257;
    const float* r2 = lwc + (i2 * 16 + ln) * 257;
    v2f a1; a1[0] = r1[k]; a1[1] = r1[k + 1];
    v2f a2; a2[0] = r2[k]; a2[1] = r2[k + 1];
    v2f bb; bb[0] = lm[k * 65 + j * 16 + ln]; bb[1] = lm[(k + 1) * 65 + j * 16 + ln];
    acc1 = wmma4(a1, bb, acc1);
    acc2 = wmma4(a2, bb, acc2);
  }
  float* Pb = P + b * 4096;
  for (int r = 0; r < 8; ++r) {
    int r1 = i1 * 16 + half * 8 + r, r2 = i2 * 16 + half * 8 + r, cc = j * 16 + ln;
    lp[r1 * 65 + cc] = acc1[r]; Pb[r1 * 64 + cc] = acc1[r];
    lp[r2 * 65 + cc] = acc2[r]; Pb[r2 * 64 + cc] = acc2[r];
  }
  __syncthreads();
  if (t < 64) {
    const float* pr = lp + t * 65;
    float ys = 0.0f;
    for (int c = 0; c < 64; ++c) ys += pr[c] * lsx[c];
    float ss = 0.0f;
    for (int c1 = 0; c1 < 64; ++c1) {
      float u = 0.0f;
      const float* gr = lg + c1 * 65;
      for (int c2 = 0; c2 < 64; ++c2) u += gr[c2] * pr[c2];
      ss += pr[c1] * u;
    }
    ysum[b * 64 + t] = ys;
    yssum[b * 64 + t] = ss;
  }
}

// --------------------------------------------------------------------------
// BN1 moments -> a1, c1; tv = Wl @ c1.
__global__ void k_bn1(const float* __restrict__ ysum, const float* __restrict__ yssum,
                      const float* __restrict__ g1, const float* __restrict__ b1,
                      const float* __restrict__ wl, float* __restrict__ a1o,
                      float* __restrict__ c1o, float* __restrict__ tvo) {
  __shared__ float lc1[64];
  const int t = threadIdx.x;
  if (t < 64) {
    float s1 = 0.0f, s2 = 0.0f;
    for (int b = 0; b < NB; ++b) { s1 += ysum[b * 64 + t]; s2 += yssum[b * 64 + t]; }
    float mean = s1 * BLF;
    float var = s2 * BLF - mean * mean;
    float a = g1[t] * rsqrtf(var + EPS_);
    float c = b1[t] - a * mean;
    lc1[t] = c; a1o[t] = a; c1o[t] = c;
  }
  __syncthreads();
  if (t < 64) {
    float s = 0.0f;
    for (int c = 0; c < 64; ++c) s += wl[t * 64 + c] * lc1[c];
    tvo[t] = s;
  }
}

// --------------------------------------------------------------------------
// Q_b = Wl @ (diag(a1) P_b + I); wsum_b = Q sx; wssum_b = qGq + 2 t (q.sx).
__global__ void k_qmat(const float* __restrict__ wl, const float* __restrict__ P,
                       const float* __restrict__ G, const float* __restrict__ SX,
                       const float* __restrict__ a1, const float* __restrict__ tv,
                       float* __restrict__ Q, float* __restrict__ wsum,
                       float* __restrict__ wssum) {
  extern __shared__ float sm[];
  float* lwl = sm;               // 64 x 65
  float* lbm = lwl + 64 * 65;    // 64 x 65
  float* lq  = lbm + 64 * 65;    // 64 x 65
  float* lg  = lq + 64 * 65;     // 64 x 65
  float* lsx = lg + 64 * 65;     // 64
  const int t = threadIdx.x, b = blockIdx.x;
  for (int i = t; i < 4096; i += 256) {
    int r = i >> 6, c = i & 63;
    lwl[r * 65 + c] = wl[i];
    float bm = a1[r] * P[b * 4096 + i];
    if (r == c) bm += 1.0f;
    lbm[r * 65 + c] = bm;
    lg[r * 65 + c] = G[b * 4096 + i];
  }
  if (t < 64) lsx[t] = SX[b * 64 + t];
  __syncthreads();
  for (int i = t; i < 4096; i += 256) {
    int o = i >> 6, n = i & 63;
    float s = 0.0f;
    for (int c = 0; c < 64; ++c) s += lwl[o * 65 + c] * lbm[c * 65 + n];
    lq[o * 65 + n] = s;
    Q[b * 4096 + i] = s;
  }
  __syncthreads();
  if (t < 64) {
    const float* qr = lq + t * 65;
    float qs = 0.0f;
    for (int c = 0; c < 64; ++c) qs += qr[c] * lsx[c];
    float ss = 0.0f;
    for (int c1 = 0; c1 < 64; ++c1) {
      float u = 0.0f;
      const float* gr = lg + c1 * 65;
      for (int c2 = 0; c2 < 64; ++c2) u += gr[c2] * qr[c2];
      ss += qr[c1] * u;
    }
    wsum[b * 64 + t] = qs;
    wssum[b * 64 + t] = ss + 2.0f * tv[t] * qs;
  }
}

// --------------------------------------------------------------------------
// BN2 moments -> a2, k; R_b = diag(a2) Q_b.
__global__ void k_bn2(const float* __restrict__ wsum, const float* __restrict__ wssum,
                      const float* __restrict__ g2, const float* __restrict__ b2,
                      const float* __restrict__ tv, const float* __restrict__ Q,
                      float* __restrict__ R, float* __restrict__ kv) {
  __shared__ float la2[64];
  const int t = threadIdx.x;
  if (t < 64) {
    float s1 = 0.0f, s2 = 0.0f;
    for (int b = 0; b < NB; ++b) { s1 += wsum[b * 64 + t]; s2 += wssum[b * 64 + t]; }
    float tt = tv[t];
    float mean = s1 * BLF + tt;
    float e2 = s2 * BLF + tt * tt;
    float var = e2 - mean * mean;
    float a = g2[t] * rsqrtf(var + EPS_);
    la2[t] = a;
    kv[t] = a * (tt - mean) + b2[t];
  }
  __syncthreads();
  for (int i = t; i < NB * 4096; i += 256) {
    int o = (i >> 6) & 63;
    R[i] = la2[o] * Q[i];
  }
}

// --------------------------------------------------------------------------
// Pass 2: out = relu(R_b @ x_b + k). grid = 16*32, block = 256.
__global__ void k_apply(const float* __restrict__ x, const float* __restrict__ R,
                        const float* __restrict__ kv, float* __restrict__ out) {
  extern __shared__ float sm[];
  float* lx = sm;              // 64 x XS  (MUST stay first: TDM writes LDS offset 0)
  float* lr = lx + 64 * XS;    // 64 x 65
  float* lk = lr + 64 * 65;    // 64
  const int t = threadIdx.x;
  const int b = blockIdx.x >> 5;
  const int l0 = (blockIdx.x & 31) * 256;
  const float* xb = x + ((size_t)b * CIN) * LSTRIDE + l0;
  if (t == 0) __builtin_prefetch(R + b * 4096, 0, 1);  // global_prefetch_b8
#if HAVE_TDM
  if (t == 0) {
    tdm_load_x_tile(xb);
  }
#else
  for (int i = t; i < 64 * 64; i += 256) {
    int row = i >> 6, c4 = i & 63;
    float4 v = reinterpret_cast<const float4*>(xb + (size_t)row * LSTRIDE)[c4];
    float* d = lx + row * XS + c4 * 4;
    d[0] = v.x; d[1] = v.y; d[2] = v.z; d[3] = v.w;
  }
#endif
  for (int i = t; i < 4096; i += 256) lr[(i >> 6) * 65 + (i & 63)] = R[b * 4096 + i];
  if (t < 64) lk[t] = kv[t];
#if HAVE_TDM
  if (t < 32) __builtin_amdgcn_s_wait_tensorcnt(0);
#endif
  __syncthreads();
  const int wave = t >> 5, lane = t & 31, half = lane >> 4, ln = lane & 15;
  const int mt = wave & 3;           // output-channel tile
  const int nb = (wave >> 2) * 8;    // 8 l-tiles per wave
  v8f acc[8] = {};
  for (int k0 = 0; k0 < 64; k0 += 4) {
    const int k = k0 + 2 * half;
    const float* ra = lr + (mt * 16 + ln) * 65;
    v2f a; a[0] = ra[k]; a[1] = ra[k + 1];
    for (int s = 0; s < 8; ++s) {
      const int c0 = (nb + s) * 16;
      v2f bb; bb[0] = lx[k * XS + c0 + ln]; bb[1] = lx[(k + 1) * XS + c0 + ln];
      acc[s] = wmma4(a, bb, acc[s]);
    }
  }
  __syncthreads();  // everyone done reading lx; reuse it as the output tile
  for (int s = 0; s < 8; ++s) {
    const int col = (nb + s) * 16 + ln;
    for (int r = 0; r < 8; ++r) {
      const int row = mt * 16 + half * 8 + r;
      lx[row * XS + col] = fmaxf(acc[s][r] + lk[row], 0.0f);
    }
  }
  __syncthreads();
  float* ob = out + ((size_t)b * CIN) * LSTRIDE + l0;
  for (int i = t; i < 64 * 64; i += 256) {   // coalesced float4 stores
    int row = i >> 6, c4 = i & 63;
    const float* sp = lx + row * XS + c4 * 4;
    float4 v; v.x = sp[0]; v.y = sp[1]; v.z = sp[2]; v.w = sp[3];
    reinterpret_cast<float4*>(ob + (size_t)row * LSTRIDE)[c4] = v;
  }
}

// --------------------------------------------------------------------------
extern "C" void kernel_launch(void* const* d_in, const int* in_sizes, int n_in,
                              void* d_out, int out_size, void* d_ws, size_t ws_size,
                              hipStream_t stream) {
  (void)in_sizes; (void)n_in; (void)out_size; (void)ws_size;
  const float* x  = (const float*)d_in[0];
  const float* wk = (const float*)d_in[1];
  const float* wq = (const float*)d_in[2];
  const float* wv = (const float*)d_in[3];
  const float* wc = (const float*)d_in[4];
  const float* g1 = (const float*)d_in[5];
  const float* b1 = (const float*)d_in[6];
  const float* wl = (const float*)d_in[7];
  const float* g2 = (const float*)d_in[8];
  const float* b2 = (const float*)d_in[9];
  float* out = (float*)d_out;
  float* ws = (float*)d_ws;

  float* G     = ws;               // 16*64*64
  float* SX    = G + 65536;        // 16*64
  float* M     = SX + 1024;        // 16*256*64
  float* T     = M + 262144;       // 16*256*64
  float* P     = T + 262144;       // 16*64*64
  float* ysum  = P + 65536;        // 16*64
  float* yssum = ysum + 1024;      // 16*64
  float* a1    = yssum + 1024;     // 64
  float* c1    = a1 + 64;          // 64
  float* tv    = c1 + 64;          // 64
  float* Q     = tv + 64;          // 16*64*64
  float* wsum  = Q + 65536;        // 16*64
  float* wssum = wsum + 1024;      // 16*64
  float* R     = wssum + 1024;     // 16*64*64
  float* kv    = R + 65536;        // 64

  const int nzero = 65536 + 1024 + 262144;  // G, SX, M contiguous
  k_zero<<<(nzero + 255) / 256, 256, 0, stream>>>(G, nzero);

  size_t lds;
  lds = (size_t)(64 * XS) * sizeof(float);
  k_gram<<<512, 256, lds, stream>>>(x, G, SX);

  lds = (size_t)(256 * 65 + 64 * 65) * sizeof(float);
  k_tmat<<<16, 256, lds, stream>>>(wq, G, T);

  lds = (size_t)(256 * 65 + 16 * 65 + 256 * 17 + 16 * 17 + 16 + 16 + 16 * 65) * sizeof(float);
  k_scores<<<256, 256, lds, stream>>>(wk, wv, T, M);

  lds = (size_t)(64 * 257 + 256 * 65 + 64 * 65 + 64 * 65 + 64) * sizeof(float);
  k_pmat<<<16, 256, lds, stream>>>(wc, M, G, SX, P, ysum, yssum);

  k_bn1<<<1, 256, 0, stream>>>(ysum, yssum, g1, b1, wl, a1, c1, tv);

  lds = (size_t)(4 * 64 * 65 + 64) * sizeof(float);
  k_qmat<<<16, 256, lds, stream>>>(wl, P, G, SX, a1, tv, Q, wsum, wssum);

  k_bn2<<<1, 256, 0, stream>>>(wsum, wssum, g2, b2, tv, Q, R, kv);

  lds = (size_t)(64 * XS + 64 * 65 + 64) * sizeof(float);
  k_apply<<<512, 256, lds, stream>>>(x, R, kv, out);
}